// JointNet_23338852286935
// MI455X (gfx1250) — compile-verified
//
#include <hip/hip_runtime.h>
#include <hip/hip_bf16.h>

// Problem constants (match reference)
#define B_ 4
#define T_ 256
#define U_ 65
#define D_ 512
#define V_ 1024

typedef __attribute__((ext_vector_type(2))) float v2f;
typedef __attribute__((ext_vector_type(4))) float v4f;
typedef __attribute__((ext_vector_type(8))) float v8f;

// ---------------------------------------------------------------------------
// Kernel 0: tanh-approx GELU, elementwise (matches jax.nn.gelu approximate=True)
// ---------------------------------------------------------------------------
__device__ __forceinline__ float gelu_tanh(float x) {
    const float k0 = 0.7978845608028654f;   // sqrt(2/pi)
    const float k1 = 0.044715f;
    float inner = k0 * (x + k1 * x * x * x);
    return 0.5f * x * (1.0f + tanhf(inner));
}

__global__ void __launch_bounds__(256)
gelu_kernel(const float* __restrict__ in, float* __restrict__ out, int n) {
    int i = blockIdx.x * blockDim.x + threadIdx.x;
    if (i < n) out[i] = gelu_tanh(in[i]);
}

// ---------------------------------------------------------------------------
// Kernel 1: C[M x N] = A[M x K] * W[:, colOff : colOff+K]^T   (fp32 WMMA)
//   A: row-major M x K (K = D_ = 512)
//   W: row-major V x 2D (row stride WS = 1024); B-matrix element B[k][n] = W[n][colOff+k]
//   Per wave: one 16(M) x 64(N) tile, 4 f32 accumulators, K-loop step 4.
//   Fragment layouts per CDNA5 ISA 7.12.2 (32-bit A 16x4, B 4x16, C/D 16x16):
//     A: lane<16 -> M=lane,  K={k,k+1}; lane>=16 -> M=lane-16, K={k+2,k+3}
//     B: lane<16 -> N=lane,  K={k,k+1}; lane>=16 -> N=lane-16, K={k+2,k+3}
//     C: VGPR r -> row = r + 8*(lane>=16), col = lane&15
// ---------------------------------------------------------------------------
__global__ void __launch_bounds__(128)
wmma_gemm_f32(const float* __restrict__ A, const float* __restrict__ W,
              float* __restrict__ C, int Mreal, int colOff) {
    const int K  = D_;        // 512
    const int N  = V_;        // 1024
    const int WS = 2 * D_;    // 1024 (W row stride)

    const int lane  = threadIdx.x & 31;
    const int wave  = threadIdx.x >> 5;
    const int m0    = blockIdx.y * 64 + wave * 16;   // 4 waves stacked in M
    const int n0    = blockIdx.x * 64;

    const int lhalf = lane >> 4;      // 0 or 1 (selects K pair)
    const int l15   = lane & 15;

    int mA = m0 + l15;
    if (mA >= Mreal) mA = Mreal - 1;  // clamp: OOB rows compute garbage, never stored

    const float* Arow = A + (size_t)mA * K + (lhalf << 1);
    const float* Wb   = W + colOff + (lhalf << 1) + (size_t)(n0 + l15) * WS;
    const float* W0   = Wb;
    const float* W1   = Wb + (size_t)16 * WS;
    const float* W2   = Wb + (size_t)32 * WS;
    const float* W3   = Wb + (size_t)48 * WS;

    v8f acc0 = {}, acc1 = {}, acc2 = {}, acc3 = {};

    for (int k = 0; k < K; k += 4) {
        v2f a  = *(const v2f*)(Arow + k);
        v2f b0 = *(const v2f*)(W0 + k);
        v2f b1 = *(const v2f*)(W1 + k);
        v2f b2 = *(const v2f*)(W2 + k);
        v2f b3 = *(const v2f*)(W3 + k);
        // 8 args: (neg_a, A, neg_b, B, c_mod, C, reuse_a, reuse_b)
        acc0 = __builtin_amdgcn_wmma_f32_16x16x4_f32(false, a, false, b0, (short)0, acc0, false, false);
        acc1 = __builtin_amdgcn_wmma_f32_16x16x4_f32(false, a, false, b1, (short)0, acc1, false, false);
        acc2 = __builtin_amdgcn_wmma_f32_16x16x4_f32(false, a, false, b2, (short)0, acc2, false, false);
        acc3 = __builtin_amdgcn_wmma_f32_16x16x4_f32(false, a, false, b3, (short)0, acc3, false, false);
    }

    const int rowBase = m0 + (lhalf << 3);
    const int col     = n0 + l15;
#pragma unroll
    for (int r = 0; r < 8; ++r) {
        int row = rowBase + r;
        if (row < Mreal) {
            float* cp = C + (size_t)row * N + col;
            cp[0]  = acc0[r];
            cp[16] = acc1[r];
            cp[32] = acc2[r];
            cp[48] = acc3[r];
        }
    }
}

// ---------------------------------------------------------------------------
// Kernel 2: out[b,t,u,v] = encP[b,t,v] + decP[b,u,v]
//   One block per (b,t). Each thread keeps its float4 of the enc row in
//   registers across the whole u-loop; output stored non-temporally
//   (272 MB write-once stream — keep it out of L2).
// ---------------------------------------------------------------------------
__global__ void __launch_bounds__(256)
bcast_add_kernel(const float* __restrict__ encP, const float* __restrict__ decP,
                 float* __restrict__ out) {
    const int bt = blockIdx.x;          // 0 .. B*T-1
    const int b  = bt / T_;
    const int v4 = threadIdx.x << 2;    // 0..1020, V_ = 256 threads * 4

    v4f e = *(const v4f*)(encP + (size_t)bt * V_ + v4);
    const float* dbase = decP + (size_t)b * U_ * V_ + v4;
    float*       obase = out  + (size_t)bt * U_ * V_ + v4;

    for (int u = 0; u < U_; ++u) {
        v4f d = *(const v4f*)(dbase + (size_t)u * V_);
        v4f r = e + d;
        __builtin_nontemporal_store(r, (v4f*)(obase + (size_t)u * V_));
    }
}

// ---------------------------------------------------------------------------
// Host-side launcher
// ---------------------------------------------------------------------------
extern "C" void kernel_launch(void* const* d_in, const int* in_sizes, int n_in,
                              void* d_out, int out_size, void* d_ws, size_t ws_size,
                              hipStream_t stream) {
    const float* enc = (const float*)d_in[0];   // (B,T,D)
    const float* dec = (const float*)d_in[1];   // (B,U,D)
    const float* w   = (const float*)d_in[2];   // (V, 2D)
    float*       out = (float*)d_out;           // (B,T,U,V)

    // Workspace layout (all 16B aligned): ge | gd | enc_proj | dec_proj  (~7.6 MB)
    float* ge   = (float*)d_ws;                 // B*T*D
    float* gd   = ge   + (size_t)B_ * T_ * D_;  // B*U*D
    float* encP = gd   + (size_t)B_ * U_ * D_;  // B*T*V
    float* decP = encP + (size_t)B_ * T_ * V_;  // B*U*V

    const int nEnc = B_ * T_ * D_;   // 524288
    const int nDec = B_ * U_ * D_;   // 133120

    gelu_kernel<<<(nEnc + 255) / 256, 256, 0, stream>>>(enc, ge, nEnc);
    gelu_kernel<<<(nDec + 255) / 256, 256, 0, stream>>>(dec, gd, nDec);

    // enc projection: M = B*T = 1024 rows, N = V = 1024, weight cols [0, D)
    {
        dim3 grid(V_ / 64, (B_ * T_) / 64);     // (16, 16)
        wmma_gemm_f32<<<grid, 128, 0, stream>>>(ge, w, encP, B_ * T_, 0);
    }
    // dec projection: M = B*U = 260 rows (ragged), weight cols [D, 2D)
    {
        dim3 grid(V_ / 64, (B_ * U_ + 63) / 64); // (16, 5)
        wmma_gemm_f32<<<grid, 128, 0, stream>>>(gd, w, decP, B_ * U_, D_);
    }

    // Broadcast add -> 272.6 MB output stream (the true bottleneck)
    bcast_add_kernel<<<B_ * T_, 256, 0, stream>>>(encP, decP, out);
}